// TtMambaSSM_35862976922071
// MI455X (gfx1250) — compile-verified
//
#include <hip/hip_runtime.h>
#include <hip/hip_bf16.h>

// Problem dims (fixed by the reference).
#define U_DIM 512
#define E_DIM 5120
#define N_DIM 16
#define R_DIM 320          // E/N
#define EN_DIM (E_DIM * N_DIM)   // 81920

typedef __attribute__((ext_vector_type(2))) float v2f;
typedef __attribute__((ext_vector_type(8))) float v8f;

// ---------------------------------------------------------------------------
// One-wave WMMA f32 GEMM: each 32-thread block computes a 16x16 tile of
// C = A(MxK) * B(KxN) using V_WMMA_F32_16X16X4_F32 (exact f32 — matches the
// reference's f32 math).  Optional per-column bias + softplus on store.
//
// Operand layouts per CDNA5 ISA 7.12.2 (wave32):
//   A 16x4:  lane L -> row M=L&15, holds K = k+2*(L>>4), k+2*(L>>4)+1
//   B 4x16:  lane L -> col N=L&15, holds K rows k+2*(L>>4), +1
//   C/D:     lane L, VGPR v -> D[ M = v + 8*(L>>4) ][ N = L&15 ]
// ---------------------------------------------------------------------------
__global__ __launch_bounds__(32)
void wmma_gemm_f32(const float* __restrict__ A, const float* __restrict__ B,
                   const float* __restrict__ bias, float* __restrict__ C,
                   int M, int N, int K, int do_softplus) {
    const int lane = threadIdx.x & 31;
    const int l15  = lane & 15;
    const int half = lane >> 4;          // 0 or 1
    const int tm   = blockIdx.y * 16;
    const int tn   = blockIdx.x * 16;
    const int am   = tm + l15;           // A row this lane loads
    const int bn   = tn + l15;           // B column this lane loads / stores

    const float* __restrict__ Arow = A + (size_t)am * K;

    v8f acc = {};
    for (int k = 0; k < K; k += 4) {
        const int ka = k + half * 2;
        // A: contiguous float2 within this lane's row.
        v2f a = *(const v2f*)(Arow + ka);
        // B: two rows of B, same column.
        v2f b;
        b.x = B[(size_t)ka       * N + bn];
        b.y = B[(size_t)(ka + 1) * N + bn];
        acc = __builtin_amdgcn_wmma_f32_16x16x4_f32(
            /*neg_a=*/false, a, /*neg_b=*/false, b,
            /*c_mod=*/(short)0, acc, /*reuse_a=*/false, /*reuse_b=*/false);
    }

    float bv = bias ? bias[bn] : 0.0f;
#pragma unroll
    for (int v = 0; v < 8; ++v) {
        const int row = tm + half * 8 + v;
        float z = acc[v] + bv;
        if (do_softplus) {
            // numerically stable softplus: max(z,0) + log1p(exp(-|z|))
            z = fmaxf(z, 0.0f) + log1pf(expf(-fabsf(z)));
        }
        C[(size_t)row * N + bn] = z;
    }
}

// ---------------------------------------------------------------------------
// Elementwise SSM state update over U x (E*16) with the reference's mixed
// broadcast layouts:
//   h_new[u,j] = exp(dt[u, j>>4] * A[u,j]) * h[u,j]
//              + dt[u, j>>4] * B0[u, j/5120] * x[u, j>>4]
// float4-vectorized: a 4-aligned group never crosses a j>>4 or j/5120 boundary.
// ---------------------------------------------------------------------------
__global__ __launch_bounds__(256)
void ssm_update_kernel(const float* __restrict__ x, const float* __restrict__ h,
                       const float* __restrict__ A, const float* __restrict__ dt,
                       const float* __restrict__ B0, float* __restrict__ hnew) {
    const unsigned idx = blockIdx.x * blockDim.x + threadIdx.x;   // one float4
    const unsigned i   = idx * 4u;                                // flat element
    const unsigned u   = i / (unsigned)EN_DIM;
    const unsigned j   = i - u * (unsigned)EN_DIM;
    const unsigned e   = j >> 4;                 // repeat-by-16 index (dt, x)
    const unsigned chk = j / (unsigned)E_DIM;    // repeat-by-E index (B0)

    const float dtv = dt[u * (unsigned)E_DIM + e];
    const float xv  = x [u * (unsigned)E_DIM + e];
    const float bb  = B0[u * (unsigned)N_DIM + chk];
    const float c   = dtv * bb * xv;

    const float4 Av = ((const float4*)A)[idx];
    const float4 hv = ((const float4*)h)[idx];

    float4 o;
    o.x = expf(dtv * Av.x) * hv.x + c;
    o.y = expf(dtv * Av.y) * hv.y + c;
    o.z = expf(dtv * Av.z) * hv.z + c;
    o.w = expf(dtv * Av.w) * hv.w + c;
    ((float4*)hnew)[idx] = o;
}

// out = x * D + x, float4-vectorized.
__global__ __launch_bounds__(256)
void out_kernel(const float* __restrict__ x, const float* __restrict__ D,
                float* __restrict__ out) {
    const unsigned idx = blockIdx.x * blockDim.x + threadIdx.x;
    const float4 xv = ((const float4*)x)[idx];
    const float4 dv = ((const float4*)D)[idx];
    float4 o;
    o.x = xv.x * dv.x + xv.x;
    o.y = xv.y * dv.y + xv.y;
    o.z = xv.z * dv.z + xv.z;
    o.w = xv.w * dv.w + xv.w;
    ((float4*)out)[idx] = o;
}

extern "C" void kernel_launch(void* const* d_in, const int* in_sizes, int n_in,
                              void* d_out, int out_size, void* d_ws, size_t ws_size,
                              hipStream_t stream) {
    (void)in_sizes; (void)n_in; (void)out_size; (void)ws_size;

    const float* x    = (const float*)d_in[0];  // (U,E)
    // d_in[1] = h (U,E*16)
    const float* h    = (const float*)d_in[1];
    const float* W1   = (const float*)d_in[2];  // (E,R) delta_t_proj_w
    const float* W2   = (const float*)d_in[3];  // (R,E) dt_proj_w
    const float* bias = (const float*)d_in[4];  // (E,)  dt_proj_b
    const float* Bw   = (const float*)d_in[5];  // (E,N) B_proj_w
    const float* A    = (const float*)d_in[6];  // (U,E*16)
    // d_in[7] = C_proj_w -> dead code in the reference, skipped entirely.
    const float* D    = (const float*)d_in[8];  // (U,E)

    float* out  = (float*)d_out;                       // (U,E)
    float* hnew = out + (size_t)U_DIM * E_DIM;         // (U,E*16)

    float* T  = (float*)d_ws;                          // (U,R)
    float* dt = T + (size_t)U_DIM * R_DIM;             // (U,E)
    float* B0 = dt + (size_t)U_DIM * E_DIM;            // (U,N)

    // 1) T = x @ W1               (512x320, K=5120)
    wmma_gemm_f32<<<dim3(R_DIM / 16, U_DIM / 16), 32, 0, stream>>>(
        x, W1, nullptr, T, U_DIM, R_DIM, E_DIM, 0);

    // 2) dt = softplus(T @ W2 + b) (512x5120, K=320)
    wmma_gemm_f32<<<dim3(E_DIM / 16, U_DIM / 16), 32, 0, stream>>>(
        T, W2, bias, dt, U_DIM, E_DIM, R_DIM, 1);

    // 3) B0 = x @ B_proj_w         (512x16, K=5120)
    wmma_gemm_f32<<<dim3(1, U_DIM / 16), 32, 0, stream>>>(
        x, Bw, nullptr, B0, U_DIM, N_DIM, E_DIM, 0);

    // 4) h_new state update (HBM-bound: ~500 MB of traffic, the real cost)
    {
        const unsigned total4 = (unsigned)U_DIM * EN_DIM / 4u;   // 10,485,760
        ssm_update_kernel<<<total4 / 256, 256, 0, stream>>>(x, h, A, dt, B0, hnew);
    }

    // 5) out = x*D + x
    {
        const unsigned n4 = (unsigned)U_DIM * E_DIM / 4u;        // 655,360
        out_kernel<<<n4 / 256, 256, 0, stream>>>(x, D, out);
    }
}